// CArapEnergy_46059229282938
// MI455X (gfx1250) — compile-verified
//
#include <hip/hip_runtime.h>

// ---- problem constants (from reference) ----
#define B_      16
#define N_      20000
#define K_      16
#define LATENT_ 256
#define HIDDEN_ 1024
#define M3_     (N_ * 3)     // 60000

// backward GEMM K-split: 60 chunks of 1000 over the 60000 dim
#define BWD_KSPLIT  60
#define BWD_KCHUNK  (M3_ / BWD_KSPLIT)   // 1000 (multiple of 4)

typedef float v2f __attribute__((ext_vector_type(2)));
typedef float v8f __attribute__((ext_vector_type(8)));

// =====================================================================
// Kernel 1: h = relu(code @ W1 + b1)   [16,256]@[256,1024]
// =====================================================================
__global__ __launch_bounds__(256) void k_mlp1(const float* __restrict__ code,
                                              const float* __restrict__ W1,
                                              const float* __restrict__ b1,
                                              float* __restrict__ h) {
    int tid = blockIdx.x * blockDim.x + threadIdx.x;   // 16*1024 threads
    int bb = tid / HIDDEN_;
    int jj = tid % HIDDEN_;
    if (bb >= B_) return;
    float acc = b1[jj];
    #pragma unroll 8
    for (int l = 0; l < LATENT_; ++l)
        acc = fmaf(code[bb * LATENT_ + l], W1[l * HIDDEN_ + jj], acc);
    h[bb * HIDDEN_ + jj] = fmaxf(acc, 0.0f);
}

// =====================================================================
// Kernel 2: recon = h @ W2 + b2 via V_WMMA_F32_16X16X4_F32
//   M = 16 (batch), N tile = 16 output cols, K loop over 1024 hidden.
//   3750 waves, each streams a 64KB column block of W2 (coalesced 64B
//   row segments per lane-half). Output stored transposed recon_t[m][b].
// =====================================================================
__global__ __launch_bounds__(256) void k_gemm_fwd(const float* __restrict__ h,
                                                  const float* __restrict__ W2,
                                                  const float* __restrict__ b2,
                                                  float* __restrict__ recon_t) {
    int wave = blockIdx.x * (blockDim.x >> 5) + (threadIdx.x >> 5);
    if (wave >= M3_ / 16) return;                      // 3750 tiles
    int n0   = wave * 16;
    int lane = threadIdx.x & 31;
    int l16  = lane & 15;
    int half = lane >> 4;

    const float* hrow = h + l16 * HIDDEN_ + half * 2;                 // A row m=l16
    const float* wcol = W2 + (size_t)(half * 2) * M3_ + n0 + l16;     // B col n

    v8f acc = {};
    #pragma unroll 4
    for (int kk = 0; kk < HIDDEN_; kk += 4) {
        v2f a;
        a.x = hrow[kk];
        a.y = hrow[kk + 1];
        v2f b;
        b.x = wcol[(size_t)kk * M3_];
        b.y = wcol[(size_t)(kk + 1) * M3_];
        acc = __builtin_amdgcn_wmma_f32_16x16x4_f32(
            false, a, false, b, (short)0, acc, false, false);
    }

    int n = n0 + l16;
    float bias = b2[n];
    #pragma unroll
    for (int r = 0; r < 8; ++r) {
        int m = r + half * 8;                          // batch index
        recon_t[(size_t)n * 16 + m] = acc[r] + bias;
    }
}

// =====================================================================
// 3x3 SVD -> proper rotation R = V * U^T (Jacobi on S^T S)
// =====================================================================
__device__ __forceinline__ void jacobi_rot(float A[3][3], float V[3][3],
                                           int p, int q) {
    float apq = A[p][q];
    if (fabsf(apq) < 1e-30f) return;
    float tau = (A[q][q] - A[p][p]) / (2.0f * apq);
    float t   = (tau >= 0.0f ? 1.0f : -1.0f) /
                (fabsf(tau) + sqrtf(1.0f + tau * tau));
    float c = 1.0f / sqrtf(1.0f + t * t);
    float s = t * c;
    #pragma unroll
    for (int k = 0; k < 3; ++k) {
        float akp = A[k][p], akq = A[k][q];
        A[k][p] = c * akp - s * akq;
        A[k][q] = s * akp + c * akq;
    }
    #pragma unroll
    for (int k = 0; k < 3; ++k) {
        float apk = A[p][k], aqk = A[q][k];
        A[p][k] = c * apk - s * aqk;
        A[q][k] = s * apk + c * aqk;
    }
    #pragma unroll
    for (int k = 0; k < 3; ++k) {
        float vkp = V[k][p], vkq = V[k][q];
        V[k][p] = c * vkp - s * vkq;
        V[k][q] = s * vkp + c * vkq;
    }
}

__device__ __forceinline__ void svd_rotation(const float S[3][3], float R[3][3]) {
    float A[3][3];
    #pragma unroll
    for (int i = 0; i < 3; ++i)
        #pragma unroll
        for (int j = 0; j < 3; ++j)
            A[i][j] = S[0][i] * S[0][j] + S[1][i] * S[1][j] + S[2][i] * S[2][j];
    float V[3][3] = {{1.f,0.f,0.f},{0.f,1.f,0.f},{0.f,0.f,1.f}};
    #pragma unroll
    for (int sw = 0; sw < 5; ++sw) {
        jacobi_rot(A, V, 0, 1);
        jacobi_rot(A, V, 0, 2);
        jacobi_rot(A, V, 1, 2);
    }
    float ev[3] = {A[0][0], A[1][1], A[2][2]};
    #pragma unroll
    for (int i = 0; i < 2; ++i)
        #pragma unroll
        for (int j = 0; j < 2; ++j)
            if (ev[j] < ev[j + 1]) {
                float t = ev[j]; ev[j] = ev[j + 1]; ev[j + 1] = t;
                for (int k = 0; k < 3; ++k) {
                    float tv = V[k][j]; V[k][j] = V[k][j + 1]; V[k][j + 1] = tv;
                }
            }
    // right-handed V: v2 = v0 x v1
    V[0][2] = V[1][0] * V[2][1] - V[2][0] * V[1][1];
    V[1][2] = V[2][0] * V[0][1] - V[0][0] * V[2][1];
    V[2][2] = V[0][0] * V[1][1] - V[1][0] * V[0][1];

    float u0[3], u1[3], u2[3];
    #pragma unroll
    for (int i = 0; i < 3; ++i)
        u0[i] = S[i][0] * V[0][0] + S[i][1] * V[1][0] + S[i][2] * V[2][0];
    float s0 = sqrtf(u0[0]*u0[0] + u0[1]*u0[1] + u0[2]*u0[2]);
    if (s0 < 1e-12f) {
        R[0][0]=1.f;R[0][1]=0.f;R[0][2]=0.f;
        R[1][0]=0.f;R[1][1]=1.f;R[1][2]=0.f;
        R[2][0]=0.f;R[2][1]=0.f;R[2][2]=1.f;
        return;
    }
    float inv0 = 1.0f / s0;
    u0[0]*=inv0; u0[1]*=inv0; u0[2]*=inv0;
    #pragma unroll
    for (int i = 0; i < 3; ++i)
        u1[i] = S[i][0] * V[0][1] + S[i][1] * V[1][1] + S[i][2] * V[2][1];
    float dot01 = u1[0]*u0[0] + u1[1]*u0[1] + u1[2]*u0[2];
    u1[0] -= dot01*u0[0]; u1[1] -= dot01*u0[1]; u1[2] -= dot01*u0[2];
    float s1 = sqrtf(u1[0]*u1[0] + u1[1]*u1[1] + u1[2]*u1[2]);
    if (s1 > 1e-12f) {
        float inv1 = 1.0f / s1;
        u1[0]*=inv1; u1[1]*=inv1; u1[2]*=inv1;
    } else {
        float e[3] = {0.f, 0.f, 0.f};
        e[(fabsf(u0[0]) < 0.9f) ? 0 : 1] = 1.0f;
        u1[0] = u0[1]*e[2] - u0[2]*e[1];
        u1[1] = u0[2]*e[0] - u0[0]*e[2];
        u1[2] = u0[0]*e[1] - u0[1]*e[0];
        float inv1 = 1.0f / sqrtf(u1[0]*u1[0] + u1[1]*u1[1] + u1[2]*u1[2]);
        u1[0]*=inv1; u1[1]*=inv1; u1[2]*=inv1;
    }
    u2[0] = u0[1]*u1[2] - u0[2]*u1[1];
    u2[1] = u0[2]*u1[0] - u0[0]*u1[2];
    u2[2] = u0[0]*u1[1] - u0[1]*u1[0];

    #pragma unroll
    for (int i = 0; i < 3; ++i) {
        R[i][0] = V[i][0]*u0[0] + V[i][1]*u1[0] + V[i][2]*u2[0];
        R[i][1] = V[i][0]*u0[1] + V[i][1]*u1[1] + V[i][2]*u2[1];
        R[i][2] = V[i][0]*u0[2] + V[i][1]*u1[2] + V[i][2]*u2[2];
    }
}

// =====================================================================
// Kernel 3: per-(b,n) covariance, SVD rotation, energy + gradient
// scatter into grad_recon_t[m][b] (L2-resident atomics).
// =====================================================================
__global__ __launch_bounds__(256) void k_edges(const float* __restrict__ xyz1,
                                               const float* __restrict__ recon_t,
                                               const float* __restrict__ wMat,
                                               const float* __restrict__ area,
                                               const int*   __restrict__ nbrs,
                                               const int*   __restrict__ numN,
                                               float* __restrict__ grad_recon_t,
                                               float* __restrict__ outE) {
    __shared__ float sE[16];
    if (threadIdx.x < 16) sE[threadIdx.x] = 0.0f;
    __syncthreads();

    int tid = blockIdx.x * blockDim.x + threadIdx.x;   // 320000 threads
    int b = tid & 15;
    int n = tid >> 4;
    float eloc = 0.0f;

    if (n < N_) {
        int   nn = numN[n];
        float ar = area[n];
        float px = xyz1[(b * N_ + n) * 3 + 0];
        float py = xyz1[(b * N_ + n) * 3 + 1];
        float pz = xyz1[(b * N_ + n) * 3 + 2];
        float qx = recon_t[(size_t)(n * 3 + 0) * 16 + b];
        float qy = recon_t[(size_t)(n * 3 + 1) * 16 + b];
        float qz = recon_t[(size_t)(n * 3 + 2) * 16 + b];

        float S[3][3] = {{0.f,0.f,0.f},{0.f,0.f,0.f},{0.f,0.f,0.f}};
        for (int k = 0; k < K_; ++k) {
            int   j = nbrs[n * K_ + k];
            float w = (k < nn) ? wMat[n * K_ + k] : 0.0f;
            float dx = px - xyz1[(b * N_ + j) * 3 + 0];
            float dy = py - xyz1[(b * N_ + j) * 3 + 1];
            float dz = pz - xyz1[(b * N_ + j) * 3 + 2];
            float ex = qx - recon_t[(size_t)(j * 3 + 0) * 16 + b];
            float ey = qy - recon_t[(size_t)(j * 3 + 1) * 16 + b];
            float ez = qz - recon_t[(size_t)(j * 3 + 2) * 16 + b];
            S[0][0] += w * dx * ex; S[0][1] += w * dx * ey; S[0][2] += w * dx * ez;
            S[1][0] += w * dy * ex; S[1][1] += w * dy * ey; S[1][2] += w * dy * ez;
            S[2][0] += w * dz * ex; S[2][1] += w * dz * ey; S[2][2] += w * dz * ez;
        }

        float R[3][3];
        svd_rotation(S, R);

        const float invNK = 1.0f / (float)(N_ * K_);
        float gox = 0.f, goy = 0.f, goz = 0.f;
        for (int k = 0; k < K_; ++k) {
            int   j = nbrs[n * K_ + k];
            float w = (k < nn) ? wMat[n * K_ + k] : 0.0f;
            float dx = px - xyz1[(b * N_ + j) * 3 + 0];
            float dy = py - xyz1[(b * N_ + j) * 3 + 1];
            float dz = pz - xyz1[(b * N_ + j) * 3 + 2];
            float ex = qx - recon_t[(size_t)(j * 3 + 0) * 16 + b];
            float ey = qy - recon_t[(size_t)(j * 3 + 1) * 16 + b];
            float ez = qz - recon_t[(size_t)(j * 3 + 2) * 16 + b];
            float rx = ex - (R[0][0]*dx + R[0][1]*dy + R[0][2]*dz);
            float ry = ey - (R[1][0]*dx + R[1][1]*dy + R[1][2]*dz);
            float rz = ez - (R[2][0]*dx + R[2][1]*dy + R[2][2]*dz);
            float cw = ar * w * invNK;
            eloc += cw * (rx*rx + ry*ry + rz*rz);
            float g  = 2.0f * cw;
            float gx = g * rx, gy = g * ry, gz = g * rz;
            gox += gx; goy += gy; goz += gz;
            atomicAdd(&grad_recon_t[(size_t)(j * 3 + 0) * 16 + b], -gx);
            atomicAdd(&grad_recon_t[(size_t)(j * 3 + 1) * 16 + b], -gy);
            atomicAdd(&grad_recon_t[(size_t)(j * 3 + 2) * 16 + b], -gz);
        }
        atomicAdd(&grad_recon_t[(size_t)(n * 3 + 0) * 16 + b], gox);
        atomicAdd(&grad_recon_t[(size_t)(n * 3 + 1) * 16 + b], goy);
        atomicAdd(&grad_recon_t[(size_t)(n * 3 + 2) * 16 + b], goz);
    }

    atomicAdd(&sE[b], eloc);                            // ds_add_f32
    __syncthreads();
    if (threadIdx.x < 16) atomicAdd(&outE[threadIdx.x], sE[threadIdx.x]);
}

// =====================================================================
// Kernel 4: grad_h_pre[h,b] += sum_m W2[h,m] * grad_recon_t[m,b]
//   K-split over the 60000 dim: 64 tiles x 60 chunks = 3840 waves
//   (enough to saturate HBM on the second 245MB W2 pass).
//   Partial 16x16 tiles accumulated with global_atomic_add_f32 into a
//   zeroed 64KB buffer (L2-resident, ~1M atomics total).
// =====================================================================
__global__ __launch_bounds__(256) void k_gemm_bwd(const float* __restrict__ W2,
                                                  const float* __restrict__ grad_recon_t,
                                                  float* __restrict__ grad_h_pre) {
    int wid = blockIdx.x * (blockDim.x >> 5) + (threadIdx.x >> 5);
    if (wid >= (HIDDEN_ / 16) * BWD_KSPLIT) return;    // 3840 waves
    int tile  = wid & 63;                              // hidden tile (64 = 2^6)
    int chunk = wid >> 6;                              // K chunk
    int h0    = tile * 16;
    int m0    = chunk * BWD_KCHUNK;

    int lane = threadIdx.x & 31;
    int l16  = lane & 15;
    int half = lane >> 4;

    const float* arow = W2 + (size_t)(h0 + l16) * M3_ + m0 + half * 2;      // A[h, m]
    const float* bcol = grad_recon_t + (size_t)(m0 + half * 2) * 16 + l16;  // B[m, b]

    v8f acc = {};
    #pragma unroll 4
    for (int mm = 0; mm < BWD_KCHUNK; mm += 4) {
        v2f a;
        a.x = arow[mm];
        a.y = arow[mm + 1];
        v2f b;
        b.x = bcol[(size_t)mm * 16];
        b.y = bcol[(size_t)mm * 16 + 16];
        acc = __builtin_amdgcn_wmma_f32_16x16x4_f32(
            false, a, false, b, (short)0, acc, false, false);
    }

    #pragma unroll
    for (int r = 0; r < 8; ++r) {
        int hh = h0 + r + half * 8;                    // hidden index
        int bb = l16;                                  // batch index
        atomicAdd(&grad_h_pre[bb * HIDDEN_ + hh], acc[r]);
    }
}

// =====================================================================
// Kernel 4b: apply relu' mask in place: grad_h = grad_h_pre * (h > 0)
// =====================================================================
__global__ __launch_bounds__(256) void k_mask(const float* __restrict__ h,
                                              float* __restrict__ grad_h) {
    int tid = blockIdx.x * blockDim.x + threadIdx.x;   // 16384 threads
    if (tid >= B_ * HIDDEN_) return;
    float g = grad_h[tid];
    grad_h[tid] = (h[tid] > 0.0f) ? g : 0.0f;
}

// =====================================================================
// Kernel 5: grad_code[b,l] = sum_j grad_h[b,j] * W1[l,j]
// =====================================================================
__global__ __launch_bounds__(256) void k_gcode(const float* __restrict__ grad_h,
                                               const float* __restrict__ W1,
                                               float* __restrict__ out_gcode) {
    int tid = blockIdx.x * blockDim.x + threadIdx.x;   // 4096 threads
    if (tid >= B_ * LATENT_) return;
    int b = tid & 15;
    int l = tid >> 4;
    float acc = 0.0f;
    #pragma unroll 8
    for (int j = 0; j < HIDDEN_; ++j)
        acc = fmaf(grad_h[b * HIDDEN_ + j], W1[l * HIDDEN_ + j], acc);
    out_gcode[b * LATENT_ + l] = acc;
}

// =====================================================================
// launch
// =====================================================================
extern "C" void kernel_launch(void* const* d_in, const int* in_sizes, int n_in,
                              void* d_out, int out_size, void* d_ws, size_t ws_size,
                              hipStream_t stream) {
    const float* code = (const float*)d_in[0];
    const float* W1   = (const float*)d_in[1];
    const float* b1   = (const float*)d_in[2];
    const float* W2   = (const float*)d_in[3];
    const float* b2   = (const float*)d_in[4];
    const float* xyz1 = (const float*)d_in[5];
    const float* wMat = (const float*)d_in[6];
    const float* area = (const float*)d_in[7];
    const int*   nbrs = (const int*)d_in[8];
    const int*   numN = (const int*)d_in[9];
    float* out = (float*)d_out;

    char* ws = (char*)d_ws;
    float* h            = (float*)ws;                              // 64 KB
    float* recon_t      = (float*)(ws + (64 << 10));               // 3.84 MB
    float* grad_recon_t = recon_t + (size_t)M3_ * 16;              // 3.84 MB
    float* grad_h       = grad_recon_t + (size_t)M3_ * 16;         // 64 KB

    hipMemsetAsync(grad_recon_t, 0, (size_t)M3_ * 16 * sizeof(float), stream);
    hipMemsetAsync(grad_h, 0, (size_t)B_ * HIDDEN_ * sizeof(float), stream);
    hipMemsetAsync(out, 0, 16 * sizeof(float), stream);

    k_mlp1<<<(B_ * HIDDEN_) / 256, 256, 0, stream>>>(code, W1, b1, h);
    k_gemm_fwd<<<(M3_ / 16 + 7) / 8, 256, 0, stream>>>(h, W2, b2, recon_t);
    k_edges<<<(B_ * N_) / 256, 256, 0, stream>>>(xyz1, recon_t, wMat, area,
                                                 nbrs, numN, grad_recon_t, out);
    k_gemm_bwd<<<((HIDDEN_ / 16) * BWD_KSPLIT) / 8, 256, 0, stream>>>(
        W2, grad_recon_t, grad_h);
    k_mask<<<(B_ * HIDDEN_) / 256, 256, 0, stream>>>(h, grad_h);
    k_gcode<<<(B_ * LATENT_) / 256, 256, 0, stream>>>(grad_h, W1, out + 16);
}